// MessagePassingLayer_12111807774832
// MI455X (gfx1250) — compile-verified
//
#include <hip/hip_runtime.h>

#define F_IN 128
#define C 64

typedef float v2f __attribute__((ext_vector_type(2)));
typedef float v8f __attribute__((ext_vector_type(8)));

// ---- order-preserving f32 <-> u32 mapping (for exact segment-max via atomicMax) ----
__device__ __forceinline__ unsigned f32_order_encode(float f) {
  unsigned b = __float_as_uint(f);
  return (b & 0x80000000u) ? ~b : (b | 0x80000000u);
}
__device__ __forceinline__ float f32_order_decode(unsigned u) {
  unsigned b = (u & 0x80000000u) ? (u & 0x7FFFFFFFu) : ~u;
  return __uint_as_float(b);
}

// ---- init: out = bias broadcast; zero deg/attr_sum/lmax/denom ----
__global__ void k_init(float* __restrict__ out, const float* __restrict__ bias,
                       float* __restrict__ deg, float* __restrict__ asum,
                       unsigned* __restrict__ lmaxu, float* __restrict__ denom, int N) {
  int i = blockIdx.x * blockDim.x + threadIdx.x;
  if (i < N * C) out[i] = bias[i & (C - 1)];
  if (i < N) { deg[i] = 0.f; asum[i] = 0.f; lmaxu[i] = 0u; denom[i] = 0.f; }
}

// ---- per-dst degree + edge_attr sum ----
__global__ void k_degree(const int* __restrict__ ei, const float* __restrict__ eattr,
                         float* __restrict__ deg, float* __restrict__ asum, int E) {
  int e = blockIdx.x * blockDim.x + threadIdx.x;
  if (e >= E) return;
  int d = ei[E + e];
  atomicAdd(&deg[d], 1.0f);
  atomicAdd(&asum[d], eattr[e]);
}

// ---- fp32 WMMA GEMM: x[N,128] @ {W_l,W_r}[128,64] -> x_l,x_r  (wave per 16x16 tile) ----
__global__ void k_gemm(const float* __restrict__ x, const float* __restrict__ Wl,
                       const float* __restrict__ Wr, float* __restrict__ xl,
                       float* __restrict__ xr, int N) {
  int wave = threadIdx.x >> 5;
  int lane = threadIdx.x & 31;
  long long t = (long long)blockIdx.x * 8 + wave;     // tile id
  long long rowTiles = (long long)(N >> 4);           // N assumed multiple of 16
  if (t >= rowTiles * 8) return;                      // 4 col tiles x 2 matrices
  int sub   = (int)(t & 7);
  int m0    = (int)(t >> 3) << 4;
  int n0    = (sub >> 1) << 4;
  int which = sub & 1;
  const float* W = which ? Wr : Wl;
  float*       Y = which ? xr : xl;

  int half = lane >> 4;        // 0: K=0,1  1: K=2,3 (ISA 16x4 f32 A layout)
  int l15  = lane & 15;
  const float* arow = x + (long long)(m0 + l15) * F_IN + half * 2;
  const float* bcol = W + n0 + l15 + (long long)(half * 2) * C;

  v8f acc = {0.f, 0.f, 0.f, 0.f, 0.f, 0.f, 0.f, 0.f};
#pragma unroll 4
  for (int k0 = 0; k0 < F_IN; k0 += 4) {
    v2f a, b;
    a.x = arow[k0];
    a.y = arow[k0 + 1];
    b.x = bcol[(long long)k0 * C];
    b.y = bcol[(long long)(k0 + 1) * C];
    acc = __builtin_amdgcn_wmma_f32_16x16x4_f32(
        /*neg_a=*/false, a, /*neg_b=*/false, b,
        /*c_mod=*/(short)0, acc, /*reuse_a=*/false, /*reuse_b=*/false);
  }
  // C/D layout: VGPR r holds row m0 + r + half*8, col n0 + l15
  float* yrow = Y + (long long)(m0 + half * 8) * C + n0 + l15;
#pragma unroll
  for (int r = 0; r < 8; ++r) yrow[(long long)r * C] = acc[r];
}

// ---- wave-per-edge logits + segment-max (edges [0,E) real, [E,E+N) self-loops) ----
__global__ void k_logits(const int* __restrict__ ei, const float* __restrict__ eattr,
                         const float* __restrict__ xl, const float* __restrict__ xr,
                         const float* __restrict__ deg, const float* __restrict__ asum,
                         const float* __restrict__ We, const float* __restrict__ att,
                         float* __restrict__ logit, unsigned* __restrict__ lmaxu,
                         int E, int N) {
  int gid = blockIdx.x * blockDim.x + threadIdx.x;
  int w = gid >> 5, lane = gid & 31;
  if (w >= E + N) return;
  int s, d; float ea;
  if (w < E) { s = ei[w]; d = ei[E + w]; ea = eattr[w]; }
  else       { s = d = w - E; ea = asum[s] / fmaxf(deg[s], 1.0f); }

  float p = 0.f;
#pragma unroll
  for (int j = 0; j < 2; ++j) {
    int c = lane + j * 32;
    float m = xl[(long long)s * C + c] + xr[(long long)d * C + c] + ea * We[c];
    m = (m > 0.f) ? m : 0.2f * m;         // leaky_relu(0.2)
    p += m * att[c];
  }
  for (int off = 16; off > 0; off >>= 1) p += __shfl_xor(p, off, 32);
  if (lane == 0) {
    logit[w] = p;
    atomicMax(&lmaxu[d], f32_order_encode(p));
  }
}

// ---- exp + segment-sum; overwrite logit with exp value (recomputed every call) ----
__global__ void k_denom(const int* __restrict__ ei, float* __restrict__ logit,
                        const unsigned* __restrict__ lmaxu, float* __restrict__ denom,
                        int E, int N) {
  int i = blockIdx.x * blockDim.x + threadIdx.x;
  if (i >= E + N) return;
  int d = (i < E) ? ei[E + i] : (i - E);
  float lm = f32_order_decode(lmaxu[d]);
  float ex = __expf(logit[i] - lm);
  logit[i] = ex;
  atomicAdd(&denom[d], ex);
}

// ---- wave-per-edge weighted scatter: out[d] += alpha * x_l[s] ----
__global__ void k_aggregate(const int* __restrict__ ei, const float* __restrict__ exv,
                            const float* __restrict__ denom, const float* __restrict__ xl,
                            float* __restrict__ out, int E, int N) {
  int gid = blockIdx.x * blockDim.x + threadIdx.x;
  int w = gid >> 5, lane = gid & 31;
  if (w >= E + N) return;
  int s, d;
  if (w < E) { s = ei[w]; d = ei[E + w]; } else { s = d = w - E; }
  float alpha = exv[w] / denom[d];
#pragma unroll
  for (int j = 0; j < 2; ++j) {
    int c = lane + j * 32;
    atomicAdd(&out[(long long)d * C + c], alpha * xl[(long long)s * C + c]);
  }
}

extern "C" void kernel_launch(void* const* d_in, const int* in_sizes, int n_in,
                              void* d_out, int out_size, void* d_ws, size_t ws_size,
                              hipStream_t stream) {
  (void)n_in; (void)out_size; (void)ws_size;
  const float* x     = (const float*)d_in[0];
  const float* eattr = (const float*)d_in[1];
  const float* Wl    = (const float*)d_in[2];
  const float* Wr    = (const float*)d_in[3];
  const float* We    = (const float*)d_in[4];
  const float* att   = (const float*)d_in[5];
  const float* bias  = (const float*)d_in[6];
  const int*   ei    = (const int*)d_in[7];
  float*       out   = (float*)d_out;

  int N = in_sizes[0] / F_IN;
  int E = in_sizes[1];

  // carve scratch (~30 MB)
  char* ws = (char*)d_ws;
  size_t o = 0;
  auto take = [&](size_t bytes) -> void* {
    void* p = ws + o; o += (bytes + 255) & ~(size_t)255; return p;
  };
  float*    xl    = (float*)take((size_t)N * C * 4);
  float*    xr    = (float*)take((size_t)N * C * 4);
  float*    deg   = (float*)take((size_t)N * 4);
  float*    asum  = (float*)take((size_t)N * 4);
  unsigned* lmaxu = (unsigned*)take((size_t)N * 4);
  float*    denom = (float*)take((size_t)N * 4);
  float*    logit = (float*)take((size_t)(E + N) * 4);

  const int B = 256;
  long long initThr = (long long)N * C;
  k_init<<<(unsigned)((initThr + B - 1) / B), B, 0, stream>>>(out, bias, deg, asum, lmaxu, denom, N);
  k_degree<<<(unsigned)((E + B - 1) / B), B, 0, stream>>>(ei, eattr, deg, asum, E);
  k_gemm<<<(unsigned)(N >> 4), 256, 0, stream>>>(x, Wl, Wr, xl, xr, N);

  long long edges = (long long)E + N;
  long long thr   = edges * 32;  // wave per edge
  k_logits<<<(unsigned)((thr + B - 1) / B), B, 0, stream>>>(ei, eattr, xl, xr, deg, asum,
                                                            We, att, logit, lmaxu, E, N);
  k_denom<<<(unsigned)((edges + B - 1) / B), B, 0, stream>>>(ei, logit, lmaxu, denom, E, N);
  k_aggregate<<<(unsigned)((thr + B - 1) / B), B, 0, stream>>>(ei, logit, denom, xl, out, E, N);
}